// Sparsegen_linear1_472446402759
// MI455X (gfx1250) — compile-verified
//
#include <hip/hip_runtime.h>
#include <cstdint>

// sparsegen-linear projection along last dim (dim = 1024), LAM = 0.5, normalized.
//
// MI455X strategy:
//  * One wave32 per row; row register-resident (32 f32/lane). Michelot's exact
//    fixed point for tau (equivalent to the sort/cumsum reference), so the
//    kernel stays at the HBM roofline (~402 MB -> ~17 us at 23.3 TB/s).
//  * CDNA5 async data mover: global_load_async_to_lds_b128 (ASYNCcnt),
//    DOUBLE-BUFFERED: prefetch row n+1 while computing row n. Async loads
//    complete in order per wave, so `s_wait_asynccnt 8` drains exactly the
//    oldest 8 (current buffer) while the next 8 remain in flight.
//  * Streaming cache policy: input+output (384 MiB) exceed the 192 MB L2 and
//    are touched once -> th:TH_LOAD_NT on async loads, TH_STORE_NT on stores.

#define LAM            0.5f
#define DIM            1024
#define WAVES_PER_BLOCK 8
#define THREADS        256     // 8 wave32
#define ELEMS_PER_LANE 32      // DIM / 32 lanes
#define F4_PER_LANE    8       // ELEMS_PER_LANE / 4
#define ROWS_PER_WAVE  4       // persistent grid-stride factor

// Native clang vector (not HIP_vector_type) so __builtin_nontemporal_store
// accepts it; still lowers to b128 memory ops.
typedef __attribute__((ext_vector_type(4))) float f32x4;

// Low 32 bits of a generic pointer into LDS == workgroup-relative LDS byte
// offset (hardware adds the LDS base and the instruction offset).
__device__ __forceinline__ unsigned lds_byte_off(const void* p) {
  return (unsigned)(uintptr_t)p;
}

// CDNA5 async LDS load: 16 bytes per lane, ASYNCcnt-tracked, non-temporal.
__device__ __forceinline__ void async_ld_f4_nt(unsigned lds_off, const float* gaddr) {
  asm volatile("global_load_async_to_lds_b128 %0, %1, off th:TH_LOAD_NT"
               :
               : "v"(lds_off), "v"((uint64_t)(uintptr_t)gaddr)
               : "memory");
}

__device__ __forceinline__ void wait_async_le0() {
  asm volatile("s_wait_asynccnt 0x0" ::: "memory");
}
// In-order completion: with 16 outstanding, <=8 means the oldest 8 are done.
__device__ __forceinline__ void wait_async_le8() {
  asm volatile("s_wait_asynccnt 0x8" ::: "memory");
}

__device__ __forceinline__ float wave_sum_f(float v) {
#pragma unroll
  for (int m = 16; m >= 1; m >>= 1) v += __shfl_xor(v, m, 32);
  return v;  // butterfly: every lane holds the total
}

__device__ __forceinline__ int wave_sum_i(int v) {
#pragma unroll
  for (int m = 16; m >= 1; m >>= 1) v += __shfl_xor(v, m, 32);
  return v;
}

// Issue the 8 async b128 loads that stage one row into one LDS buffer.
__device__ __forceinline__ void prefetch_row(float* s_row, const float* grow, int lane) {
#pragma unroll
  for (int j = 0; j < F4_PER_LANE; ++j) {
    const int e = j * 128 + lane * 4;   // contiguous 16B per lane, coalesced
    async_ld_f4_nt(lds_byte_off(&s_row[e]), grow + e);
  }
}

// Michelot fixed point + clipped/normalized output for one LDS-resident row.
__device__ __forceinline__ void process_row(const float* s_row, float* orow, int lane) {
  float z[ELEMS_PER_LANE];
#pragma unroll
  for (int j = 0; j < F4_PER_LANE; ++j) {
    const f32x4 v = *reinterpret_cast<const f32x4*>(&s_row[j * 128 + lane * 4]);
    z[4 * j + 0] = v.x;
    z[4 * j + 1] = v.y;
    z[4 * j + 2] = v.z;
    z[4 * j + 3] = v.w;
  }

  const float R = 1.0f - LAM;

  float lsum = 0.0f;
#pragma unroll
  for (int j = 0; j < ELEMS_PER_LANE; ++j) lsum += z[j];
  float tau    = (wave_sum_f(lsum) - R) * (1.0f / (float)DIM);
  int   k_prev = DIM;

  for (int it = 0; it < DIM + 2; ++it) {   // converges in <= DIM steps; ~6 typical
    float ls = 0.0f;
    int   lc = 0;
#pragma unroll
    for (int j = 0; j < ELEMS_PER_LANE; ++j) {
      const float zz = z[j];
      const bool  gt = zz > tau;
      ls += gt ? zz : 0.0f;
      lc += gt ? 1 : 0;
    }
    const float S = wave_sum_f(ls);
    const int   C = wave_sum_i(lc);
    if (C == k_prev || C == 0) break;      // support stabilized -> tau final
    k_prev = C;
    tau    = (S - R) / (float)C;           // uniform across the wave
  }

  const float scale = 1.0f / (1.0f - LAM);
#pragma unroll
  for (int j = 0; j < F4_PER_LANE; ++j) {
    f32x4 p;
    p.x = fmaxf(z[4 * j + 0] - tau, 0.0f) * scale;
    p.y = fmaxf(z[4 * j + 1] - tau, 0.0f) * scale;
    p.z = fmaxf(z[4 * j + 2] - tau, 0.0f) * scale;
    p.w = fmaxf(z[4 * j + 3] - tau, 0.0f) * scale;
    // Output written once, never re-read: stream past L2 (TH_STORE_NT).
    __builtin_nontemporal_store(p, reinterpret_cast<f32x4*>(&orow[j * 128 + lane * 4]));
  }
}

__global__ __launch_bounds__(THREADS)
void sparsegen_linear_rows(const float* __restrict__ x,
                           float* __restrict__ out,
                           int nrows) {
  // 2 buffers per wave (double buffering): 8 waves * 2 * 4KB = 64 KB.
  __shared__ float s_buf[WAVES_PER_BLOCK * 2 * DIM];

  const int wave    = threadIdx.x >> 5;
  const int lane    = threadIdx.x & 31;
  const int gwave   = blockIdx.x * WAVES_PER_BLOCK + wave;
  const int wstride = gridDim.x * WAVES_PER_BLOCK;

  float* buf0 = &s_buf[wave * 2 * DIM];
  float* buf1 = buf0 + DIM;

  int row = gwave;
  if (row >= nrows) return;

  prefetch_row(buf0, x + (size_t)row * DIM, lane);

  int cur = 0;
  for (; row < nrows; row += wstride) {
    const int  nrow      = row + wstride;     // wave-uniform
    const bool have_next = nrow < nrows;

    float* curbuf = cur ? buf1 : buf0;
    float* nxtbuf = cur ? buf0 : buf1;

    if (have_next) {
      prefetch_row(nxtbuf, x + (size_t)nrow * DIM, lane);
      wait_async_le8();   // oldest 8 (current row) complete; next 8 in flight
    } else {
      wait_async_le0();
    }

    process_row(curbuf, out + (size_t)row * DIM, lane);
    cur ^= 1;
  }
}

extern "C" void kernel_launch(void* const* d_in, const int* in_sizes, int n_in,
                              void* d_out, int out_size, void* d_ws, size_t ws_size,
                              hipStream_t stream) {
  (void)n_in; (void)out_size; (void)d_ws; (void)ws_size;
  const float* x   = (const float*)d_in[0];
  float*       out = (float*)d_out;
  const int total  = in_sizes[0];
  const int nrows  = total / DIM;   // 49152 for (4,12,1024,1024)
  // Persistent-ish launch: ~ROWS_PER_WAVE rows per wave so double buffering
  // has work to overlap. 49152 rows -> 1536 blocks of 8 waves.
  int blocks = (nrows + WAVES_PER_BLOCK * ROWS_PER_WAVE - 1) / (WAVES_PER_BLOCK * ROWS_PER_WAVE);
  if (blocks < 1) blocks = 1;
  sparsegen_linear_rows<<<dim3(blocks), dim3(THREADS), 0, stream>>>(x, out, nrows);
}